// Cell_59038620451441
// MI455X (gfx1250) — compile-verified
//
#include <hip/hip_runtime.h>

typedef __attribute__((ext_vector_type(2))) float v2f;
typedef __attribute__((ext_vector_type(8))) float v8f;

#define D 64
#define SLOPE 0.2f
#define EPS 1e-5f

#if __has_builtin(__builtin_amdgcn_global_load_async_to_lds_b128)
#define USE_ASYNC_LDS 1
#else
#define USE_ASYNC_LDS 0
#endif

// The async-to-LDS builtin takes (AS1 v4i*, AS3 v4i*, imm offset, imm cpol)
// where v4i is the GCC-vector flavor (vector_size(16) int), per the clang
// diagnostic from the signature probe.
typedef int v4i_t __attribute__((vector_size(16)));
typedef __attribute__((address_space(1))) v4i_t* gv4i_p;
typedef __attribute__((address_space(3))) v4i_t* lv4i_p;

// ---------------------------------------------------------------------------
// WMMA GEMM:  Y[N,64] = X[N,K] (row stride ldx) @ W[K,64] + bias
// 256 threads = 8 wave32 waves; each wave computes one 16x16 tile.
// W is staged into LDS once per block via the CDNA5 async-to-LDS path,
// then the inner loop feeds V_WMMA_F32_16X16X4_F32 from LDS (B) + global (A).
// ---------------------------------------------------------------------------
__global__ __launch_bounds__(256) void k_gemm_wmma(
    const float* __restrict__ X, int ldx,
    const float* __restrict__ W,  // [K,64] row-major
    const float* __restrict__ bias,
    float* __restrict__ Y, int nrows, int K)
{
    extern __shared__ float ldsW[];          // [K,64], up to 64 KB

    // ---- stage W into LDS (all threads participate; barrier below) ----
    {
        const int nchunk = (K * 64) >> 2;    // 16-byte chunks
        for (int i = threadIdx.x; i < nchunk; i += blockDim.x) {
#if USE_ASYNC_LDS
            __builtin_amdgcn_global_load_async_to_lds_b128(
                (gv4i_p)(W + (size_t)i * 4), (lv4i_p)(ldsW + (size_t)i * 4),
                /*imm offset=*/0, /*cpol=*/0);
#else
            *(float4*)(ldsW + (size_t)i * 4) = *(const float4*)(W + (size_t)i * 4);
#endif
        }
#if USE_ASYNC_LDS
#if __has_builtin(__builtin_amdgcn_s_wait_asynccnt)
        __builtin_amdgcn_s_wait_asynccnt(0);
#else
        asm volatile("s_wait_asynccnt 0x0" ::: "memory");
#endif
#endif
        __syncthreads();
    }

    const int lane = threadIdx.x & 31;
    const int wid  = threadIdx.x >> 5;
    const int rowTile = blockIdx.x * 2 + (wid >> 2);
    const int rb = rowTile * 16;

    if (rb < nrows) {                        // wave-uniform: EXEC stays all-1s
        const int cb   = (wid & 3) * 16;
        const int lrow = rb + (lane & 15);
        const int lcol = cb + (lane & 15);
        const int khalf = (lane >> 4) << 1;  // lanes 0-15 -> K0/K1, 16-31 -> K2/K3

        v8f acc = {};
        for (int kk = 0; kk < K; kk += 4) {
            const int kb = kk + khalf;
            // A: 16x4 f32 tile, 8-byte aligned pair load from global
            v2f a = *(const v2f*)(X + (size_t)lrow * ldx + kb);
            // B: 4x16 f32 tile from LDS (rows kb, kb+1 of W)
            v2f b;
            b.x = ldsW[kb * 64 + lcol];
            b.y = ldsW[(kb + 1) * 64 + lcol];
            acc = __builtin_amdgcn_wmma_f32_16x16x4_f32(
                /*neg_a=*/false, a, /*neg_b=*/false, b,
                /*c_mod=*/(short)0, acc, /*reuse_a=*/false, /*reuse_b=*/false);
        }
        // C/D layout: lane&15 = col, row = vgpr + 8*(lane>>4)
        const float bv = bias[lcol];
        const int rowoff = (lane >> 4) << 3;
#pragma unroll
        for (int j = 0; j < 8; ++j)
            Y[(size_t)(rb + rowoff + j) * 64 + lcol] = acc[j] + bv;
    }
}

// ---------------------------------------------------------------------------
// Per-column sum / sum-of-squares over the node axis (for training-mode BN)
// ---------------------------------------------------------------------------
__global__ __launch_bounds__(256) void k_colstats(
    const float* __restrict__ Y, float* __restrict__ colsum,
    float* __restrict__ colsumsq, int n)
{
    __shared__ float ss[64], sq[64];
    const int t = threadIdx.x;
    if (t < 64) { ss[t] = 0.f; sq[t] = 0.f; }
    __syncthreads();
    const int c = t & 63;
    const int rpb = blockDim.x >> 6;         // rows per block pass = 4
    float s = 0.f, q = 0.f;
    for (int r = blockIdx.x * rpb + (t >> 6); r < n; r += gridDim.x * rpb) {
        const float v = Y[(size_t)r * 64 + c];
        s += v; q += v * v;
    }
    atomicAdd(&ss[c], s);                    // ds_add_f32
    atomicAdd(&sq[c], q);
    __syncthreads();
    if (t < 64) {
        atomicAdd(&colsum[t],   ss[t]);      // global_atomic_add_f32
        atomicAdd(&colsumsq[t], sq[t]);
    }
}

__global__ void k_bnfinal(const float* __restrict__ colsum,
                          const float* __restrict__ colsumsq,
                          const float* __restrict__ g, const float* __restrict__ beta,
                          float* __restrict__ scale, float* __restrict__ shift, int n)
{
    const int c = threadIdx.x;               // 64 threads
    const float inv_n = 1.0f / (float)n;
    const float mean = colsum[c] * inv_n;
    const float var  = colsumsq[c] * inv_n - mean * mean;   // biased, as torch
    const float sc   = g[c] * rsqrtf(var + EPS);
    scale[c] = sc;
    shift[c] = beta[c] - mean * sc;
}

// bn-affine apply + optional LeakyReLU + optional strided add, strided dst
__global__ __launch_bounds__(256) void k_bnapply(
    const float* __restrict__ Y, const float* __restrict__ scale,
    const float* __restrict__ shift,
    const float* __restrict__ add, int addStride,
    float* __restrict__ dstp, int dstStride, int leaky, int n)
{
    const int i = blockIdx.x * blockDim.x + threadIdx.x;
    if (i >= n * 64) return;
    const int r = i >> 6, c = i & 63;
    float v = Y[i] * scale[c] + shift[c];
    if (leaky) v = (v >= 0.f) ? v : SLOPE * v;
    if (add) v += add[(size_t)r * addStride + c];
    dstp[(size_t)r * dstStride + c] = v;
}

// ---------------------------------------------------------------------------
// Edge aggregation kernels
// ---------------------------------------------------------------------------
__global__ __launch_bounds__(256) void k_scatter_add(
    const float* __restrict__ X, int ldx,
    const int* __restrict__ src, const int* __restrict__ dst,
    float* __restrict__ acc, int E)
{
    const int t = blockIdx.x * blockDim.x + threadIdx.x;
    if (t >= E * 16) return;
    const int e = t >> 4, c4 = (t & 15) << 2;
    const float4 v = *(const float4*)(X + (size_t)src[e] * ldx + c4);
    float* a = acc + (size_t)dst[e] * 64 + c4;
    atomicAdd(a + 0, v.x); atomicAdd(a + 1, v.y);
    atomicAdd(a + 2, v.z); atomicAdd(a + 3, v.w);
}

__device__ __forceinline__ void atomicMaxF(float* addr, float v) {
    if (v >= 0.f) atomicMax((int*)addr, __float_as_int(v));
    else          atomicMin((unsigned int*)addr, __float_as_uint(v));
}

__global__ __launch_bounds__(256) void k_scatter_max(
    const float* __restrict__ X, int ldx,
    const int* __restrict__ src, const int* __restrict__ dst,
    float* __restrict__ acc, int E)
{
    const int t = blockIdx.x * blockDim.x + threadIdx.x;
    if (t >= E * 16) return;
    const int e = t >> 4, c4 = (t & 15) << 2;
    const float4 v = *(const float4*)(X + (size_t)src[e] * ldx + c4);
    float* a = acc + (size_t)dst[e] * 64 + c4;
    atomicMaxF(a + 0, v.x); atomicMaxF(a + 1, v.y);
    atomicMaxF(a + 2, v.z); atomicMaxF(a + 3, v.w);
}

__global__ __launch_bounds__(256) void k_degree(const int* __restrict__ dst,
                                                float* __restrict__ cnt, int E)
{
    const int e = blockIdx.x * blockDim.x + threadIdx.x;
    if (e < E) atomicAdd(&cnt[dst[e]], 1.0f);
}

__global__ __launch_bounds__(256) void k_meandiv(float* __restrict__ acc,
                                                 const float* __restrict__ cnt, int n)
{
    const int i = blockIdx.x * blockDim.x + threadIdx.x;
    if (i < n * 64) acc[i] /= fmaxf(cnt[i >> 6], 1.0f);
}

__global__ __launch_bounds__(256) void k_fillbits(unsigned* __restrict__ p,
                                                  unsigned bits, int n)
{
    const int i = blockIdx.x * blockDim.x + threadIdx.x;
    if (i < n) p[i] = bits;
}

__global__ __launch_bounds__(256) void k_maxfix(float* __restrict__ acc, int n)
{
    const int i = blockIdx.x * blockDim.x + threadIdx.x;
    if (i < n * 64) {
        const float v = acc[i];
        acc[i] = isfinite(v) ? v : 0.0f;     // zero-in-degree nodes -> 0
    }
}

// ---------------------------------------------------------------------------
// Host-side orchestration
// ---------------------------------------------------------------------------
extern "C" void kernel_launch(void* const* d_in, const int* in_sizes, int /*n_in*/,
                              void* d_out, int /*out_size*/, void* d_ws, size_t /*ws_size*/,
                              hipStream_t stream)
{
    const float* V     = (const float*)d_in[0];
    const float* Wf    = (const float*)d_in[1];   // [5,64,64]
    const float* bf    = (const float*)d_in[2];   // [5,64]
    const float* gf    = (const float*)d_in[3];
    const float* betaf = (const float*)d_in[4];
    const float* Wc    = (const float*)d_in[5];   // [256,64]
    const float* bc    = (const float*)d_in[6];
    const float* g_out = (const float*)d_in[7];
    const float* b_out = (const float*)d_in[8];
    const int*   src   = (const int*)d_in[9];
    const int*   dst   = (const int*)d_in[10];

    const int N = in_sizes[0] / D;
    const int E = in_sizes[9];
    float* out = (float*)d_out;

    // Workspace layout (~77 MB for N=50000)
    float* H        = (float*)d_ws;                 // [N,256] = s1|s2|s3|s4
    float* A        = H + (size_t)N * 256;          // [N,64] aggregation scratch
    float* Yb       = A + (size_t)N * 64;           // [N,64] gemm output pre-BN
    float* cnt      = Yb + (size_t)N * 64;          // [N]
    float* colsum   = cnt + N;                      // [64]
    float* colsumsq = colsum + 64;                  // [64]
    float* scale    = colsumsq + 64;                // [64]
    float* shift    = scale + 64;                   // [64]

    const dim3 b256(256);
    const int gElem = (N * 64 + 255) / 256;
    const int gScat = (E * 16 + 255) / 256;
    const int gE    = (E + 255) / 256;
    const int gGemm = (N + 31) / 32;

    auto run_gemm_bn = [&](const float* X, int ldx, int K,
                           const float* W, const float* bias,
                           const float* g, const float* beta,
                           const float* add, int addStride,
                           float* dstp, int dstStride, int leaky) {
        const size_t smem = (size_t)K * 64 * sizeof(float);
        k_gemm_wmma<<<gGemm, b256, smem, stream>>>(X, ldx, W, bias, Yb, N, K);
        (void)hipMemsetAsync(colsum, 0, 2 * 64 * sizeof(float), stream);
        k_colstats<<<256, b256, 0, stream>>>(Yb, colsum, colsumsq, N);
        k_bnfinal<<<1, 64, 0, stream>>>(colsum, colsumsq, g, beta, scale, shift, N);
        k_bnapply<<<gElem, b256, 0, stream>>>(Yb, scale, shift, add, addStride,
                                              dstp, dstStride, leaky, N);
    };

    // in-degree counts (same dst list for both mean aggregations)
    (void)hipMemsetAsync(cnt, 0, (size_t)N * sizeof(float), stream);
    k_degree<<<gE, b256, 0, stream>>>(dst, cnt, E);

    // s1 = fc_bn(mean_agg(V), 0)              -> H[:,0:64]
    (void)hipMemsetAsync(A, 0, (size_t)N * 64 * sizeof(float), stream);
    k_scatter_add<<<gScat, b256, 0, stream>>>(V, 64, src, dst, A, E);
    k_meandiv<<<gElem, b256, 0, stream>>>(A, cnt, N);
    run_gemm_bn(A, 64, 64, Wf + 0 * 4096, bf + 0 * 64, gf + 0 * 64, betaf + 0 * 64,
                nullptr, 0, H + 0, 256, 0);

    // s2 = V + fc_bn(sum_agg(s1), 1)          -> H[:,64:128]
    (void)hipMemsetAsync(A, 0, (size_t)N * 64 * sizeof(float), stream);
    k_scatter_add<<<gScat, b256, 0, stream>>>(H + 0, 256, src, dst, A, E);
    run_gemm_bn(A, 64, 64, Wf + 1 * 4096, bf + 1 * 64, gf + 1 * 64, betaf + 1 * 64,
                V, 64, H + 64, 256, 0);

    // s3a = fc_bn(mean_agg(s1), 2)            -> H[:,128:192]
    (void)hipMemsetAsync(A, 0, (size_t)N * 64 * sizeof(float), stream);
    k_scatter_add<<<gScat, b256, 0, stream>>>(H + 0, 256, src, dst, A, E);
    k_meandiv<<<gElem, b256, 0, stream>>>(A, cnt, N);
    run_gemm_bn(A, 64, 64, Wf + 2 * 4096, bf + 2 * 64, gf + 2 * 64, betaf + 2 * 64,
                nullptr, 0, H + 128, 256, 0);

    // s3 += fc_bn(max_agg(s2), 3)             (accumulate into H[:,128:192])
    k_fillbits<<<gElem, b256, 0, stream>>>((unsigned*)A, 0xFF800000u, N * 64); // -inf
    k_scatter_max<<<gScat, b256, 0, stream>>>(H + 64, 256, src, dst, A, E);
    k_maxfix<<<gElem, b256, 0, stream>>>(A, N);
    run_gemm_bn(A, 64, 64, Wf + 3 * 4096, bf + 3 * 64, gf + 3 * 64, betaf + 3 * 64,
                H + 128, 256, H + 128, 256, 0);

    // s4 = fc_bn(sum_agg(s2), 4) + s3         -> H[:,192:256]
    (void)hipMemsetAsync(A, 0, (size_t)N * 64 * sizeof(float), stream);
    k_scatter_add<<<gScat, b256, 0, stream>>>(H + 64, 256, src, dst, A, E);
    run_gemm_bn(A, 64, 64, Wf + 4 * 4096, bf + 4 * 64, gf + 4 * 64, betaf + 4 * 64,
                H + 128, 256, H + 192, 256, 0);

    // out = leaky(bn(H @ Wc + bc)) + V
    run_gemm_bn(H, 256, 256, Wc, bc, g_out, b_out, V, 64, out, 64, 1);
}